// EnergyFromMaps_83133386981530
// MI455X (gfx1250) — compile-verified
//
#include <hip/hip_runtime.h>
#include <hip/hip_bf16.h>
#include <math.h>

// ---------------------------------------------------------------------------
// EnergyFromMaps for MI455X (gfx1250, wave32).
// Pairwise d^2 matrix computed on the matrix pipe via V_WMMA_F32_16X16X4_F32
// (rank-4 GEMM trick: [y,x,|p|^2,1] . [-2y,-2x,1,|c|^2]^T = |p-c|^2).
// All per-pair trig removed via cos/sin addition identities on precomputed
// per-point cos(a)/sin(a); the p-norm radius uses the __powf fast path.
// ---------------------------------------------------------------------------

typedef __attribute__((ext_vector_type(2))) float v2f;
typedef __attribute__((ext_vector_type(8))) float v8f;

#define PI_F 3.14159265358979323846f
#define MAXD 32.0f
#define HMAP 256
#define WMAP 256
#define NCH  32
#define NWAVES 8

// ----------------------------- helpers -------------------------------------

__device__ __forceinline__ float bilin(const float* __restrict__ img, float y, float x) {
  y = fminf(fmaxf(y, 0.0f), (float)(HMAP - 1));
  x = fminf(fmaxf(x, 0.0f), (float)(WMAP - 1));
  float y0 = floorf(y), x0 = floorf(x);
  int iy0 = (int)y0, ix0 = (int)x0;
  int iy1 = min(iy0 + 1, HMAP - 1), ix1 = min(ix0 + 1, WMAP - 1);
  float wy = y - y0, wx = x - x0;
  float v00 = img[iy0 * WMAP + ix0];
  float v01 = img[iy0 * WMAP + ix1];
  float v10 = img[iy1 * WMAP + ix0];
  float v11 = img[iy1 * WMAP + ix1];
  return v00 * (1.0f - wy) * (1.0f - wx) + v01 * (1.0f - wy) * wx +
         v10 * wy * (1.0f - wx)          + v11 * wy * wx;
}

__device__ __forceinline__ float trilin(const float* __restrict__ m3, float y, float x,
                                        float v, float vmin, float vmax, bool cyclic) {
  float c = (v - vmin) / (vmax - vmin) * (float)NCH - 0.5f;
  int i0, i1; float cw;
  if (cyclic) {
    c = c - (float)NCH * floorf(c / (float)NCH);   // non-negative (floor) mod
    float c0 = floorf(c); cw = c - c0;
    int ic = (int)c0;
    i0 = ((ic % NCH) + NCH) % NCH;
    i1 = (i0 + 1) & (NCH - 1);
  } else {
    c = fminf(fmaxf(c, 0.0f), (float)(NCH - 1));
    float c0 = floorf(c); cw = c - c0;
    i0 = (int)c0;
    i1 = min(i0 + 1, NCH - 1);
  }
  float b0 = bilin(m3 + (size_t)i0 * (HMAP * WMAP), y, x);
  float b1 = bilin(m3 + (size_t)i1 * (HMAP * WMAP), y, x);
  return (1.0f - cw) * b0 + cw * b1;
}

// superellipse radius from cos/sin of (theta - a) directly; NORM_P = 10 chain,
// final x^-0.1 via the __powf fast-math intrinsic (v_log_f32 + v_exp_f32).
__device__ __forceinline__ float radius_cs(float cphi, float sphi, float ihw, float ihl) {
  float c = fabsf(cphi) + 1e-8f;
  float s = fabsf(sphi) + 1e-8f;
  float cp = c * ihw, sp = s * ihl;        // (|cos|/hw), (|sin|/hl)
  float c2 = cp * cp, c4 = c2 * c2, c8 = c4 * c4, c10 = c8 * c2;
  float s2 = sp * sp, s4 = s2 * s2, s8 = s4 * s4, s10 = s8 * s2;
  return __powf(c10 + s10, -0.1f);
}

__device__ __forceinline__ float wave16_max(float v) {
  v = fmaxf(v, __shfl_xor(v, 1, 32));
  v = fmaxf(v, __shfl_xor(v, 2, 32));
  v = fmaxf(v, __shfl_xor(v, 4, 32));
  v = fmaxf(v, __shfl_xor(v, 8, 32));
  return v;
}
__device__ __forceinline__ float wave16_min(float v) {
  v = fminf(v, __shfl_xor(v, 1, 32));
  v = fminf(v, __shfl_xor(v, 2, 32));
  v = fminf(v, __shfl_xor(v, 4, 32));
  v = fminf(v, __shfl_xor(v, 8, 32));
  return v;
}

// --------------------------- kernel 1: prep --------------------------------
// Normalize marks, emit packed SoA:
//   pk0 = (y, x, |p|^2, a)   pk1 = (cos a, sin a, 2/w, 2/l)   ctxM = mask
// plus per-point map energies for the pts prefix.
__global__ __launch_bounds__(256) void prep_kernel(
    const float* __restrict__ cc, const unsigned char* __restrict__ mask,
    const float* __restrict__ pem, const float* __restrict__ mwm,
    const float* __restrict__ mlm, const float* __restrict__ mam,
    float4* __restrict__ pk0, float4* __restrict__ pk1, float* __restrict__ ctxM,
    float* __restrict__ peA, float* __restrict__ weA, float* __restrict__ leA,
    float* __restrict__ aeA, float* __restrict__ arA,
    int B, int G, int P) {
  int t = blockIdx.x * blockDim.x + threadIdx.x;
  if (t >= B * G) return;
  int b = t / G, q = t - b * G;

  const float* e = cc + (size_t)t * 5;
  float y = fminf(fmaxf(e[0], 0.0f), (float)HMAP);
  float x = fminf(fmaxf(e[1], 0.0f), (float)WMAP);
  float w = fminf(fmaxf(e[2], 2.0f), 32.0f);
  float l = fminf(fmaxf(e[3], 2.0f), 32.0f);
  // floor-mod (matches jnp.mod): a in [0, pi)
  float a = e[4] - PI_F * floorf(e[4] * (1.0f / PI_F));
  a = fminf(fmaxf(a, 0.0f), PI_F);
  float cm = mask[t] ? 1.0f : 0.0f;

  float sa, ca;
  __sincosf(a, &sa, &ca);
  pk0[t] = make_float4(y, x, y * y + x * x, a);
  pk1[t] = make_float4(ca, sa, 2.0f / w, 2.0f / l);   // 1/(0.5w), 1/(0.5l)
  ctxM[t] = cm;

  if (q < P) {   // pts is the prefix of ctx per batch
    int i = b * P + q;
    float pm = cm;
    peA[i] = bilin(pem, y, x) * pm;
    weA[i] = trilin(mwm, y, x, w, 2.0f, 32.0f, false) * pm;
    leA[i] = trilin(mlm, y, x, l, 2.0f, 32.0f, false) * pm;
    aeA[i] = trilin(mam, y, x, a, 0.0f, PI_F, true) * pm;
    float area = w * l;
    arA[i] = (1.0f - 2.0f * fminf(fmaxf(area * (1.0f / 256.0f), 0.0f), 1.0f)) * pm;
  }
}

// --------------------------- kernel 2: pairwise ----------------------------
// One block = (batch b, 16-row i-tile). 8 waves stride over 16-col j-tiles.
// d^2 tile from V_WMMA_F32_16X16X4_F32; trig-free per-pair energies on VALU.
__global__ __launch_bounds__(256) void pair_kernel(
    const float4* __restrict__ pk0, const float4* __restrict__ pk1,
    const float* __restrict__ ctxM,
    float* __restrict__ ovlOut, float* __restrict__ alnOut,
    int B, int G, int P) {
  __shared__ float redO[NWAVES * 16], redA[NWAVES * 16];

  int tilesPerB = P / 16;
  int b  = blockIdx.x / tilesPerB;
  int i0 = (blockIdx.x % tilesPerB) * 16;
  int base = b * G;
  int tid = threadIdx.x;

  int wave = tid >> 5, lane = tid & 31;
  int half = lane >> 4, idx = lane & 15;
  int m0 = half * 8;

  // A fragment (16x4 f32): lane(half,idx) holds row=idx, K = 2*half + r
  // A row = [y, x, |p|^2, 1]
  float4 arow = pk0[base + i0 + idx];
  v2f afrag;
  afrag.x = half ? arow.z : arow.x;
  afrag.y = half ? 1.0f   : arow.y;

  // Hoist this lane's 8 output rows (m0..m0+7) into registers: zero LDS/global
  // traffic for row data inside the j-loop.
  float rY[8], rX[8], rA[8], rCA[8], rSA[8], rIHW[8], rIHL[8], rM[8];
#pragma unroll
  for (int r = 0; r < 8; ++r) {
    int gr = base + i0 + m0 + r;
    float4 q0 = pk0[gr];
    float4 q1 = pk1[gr];
    rY[r] = q0.x; rX[r] = q0.y; rA[r] = q0.w;
    rCA[r] = q1.x; rSA[r] = q1.y; rIHW[r] = q1.z; rIHL[r] = q1.w;
    rM[r] = ctxM[gr];
  }

  float ovlacc[8], alnacc[8];
#pragma unroll
  for (int r = 0; r < 8; ++r) { ovlacc[r] = 0.0f; alnacc[r] = 1e30f; }

  int njt = G / 16;
  for (int jt = wave; jt < njt; jt += NWAVES) {
    int n = base + jt * 16 + idx;        // this lane's ctx column point
    float4 c0 = pk0[n];                  // y, x, s, a
    float4 c1 = pk1[n];                  // cos a, sin a, 2/w, 2/l
    float cmj = ctxM[n];
    if (jt + NWAVES < njt) {             // global_prefetch_b8 for next tile
      __builtin_prefetch(&pk0[n + NWAVES * 16], 0, 1);
      __builtin_prefetch(&pk1[n + NWAVES * 16], 0, 1);
    }

    // B fragment (4x16 f32): lane holds col=idx, K = 2*half + r
    // B col = [-2y, -2x, 1, |c|^2]
    v2f bfrag;
    bfrag.x = half ? 1.0f : (-2.0f * c0.x);
    bfrag.y = half ? c0.z : (-2.0f * c0.y);

    v8f cacc = {};
    v8f d2v = __builtin_amdgcn_wmma_f32_16x16x4_f32(
        /*neg_a=*/false, afrag, /*neg_b=*/false, bfrag,
        /*c_mod=*/(short)0, cacc, /*reuse_a=*/false, /*reuse_b=*/false);

#pragma unroll
    for (int r = 0; r < 8; ++r) {
      float dy = rY[r] - c0.x;
      float dx = rX[r] - c0.y;
      float nonself = (dx != 0.0f || dy != 0.0f) ? 1.0f : 0.0f;  // exact self test
      float d2c = fmaxf(fmaxf(d2v[r], 0.0f), 1e-12f);
      float inv = rsqrtf(d2c);
      float dist = d2c * inv;            // == sqrt(max(d2,1e-12))
      float pv = rM[r] * cmj;
      float dm = (pv > 0.0f) ? dist : MAXD;

      // cos/sin of theta without atan2: cosT = dy/r, sinT = dx/r
      float ct = dy * inv, st = dx * inv;
      // phi_i = theta - a_i ; phi_j = theta - a_j  (addition identities)
      float cpi = ct * rCA[r] + st * rSA[r];
      float spi = st * rCA[r] - ct * rSA[r];
      float ri = radius_cs(cpi, spi, rIHW[r], rIHL[r]);
      float cpj = ct * c1.x + st * c1.y;
      float spj = st * c1.x - ct * c1.y;
      float rj = radius_cs(cpj, spj, c1.z, c1.w);

      float ov = 1.0f - dm / (ri + rj + 1e-8f);
      ov = fminf(fmaxf(ov, 0.0f), 1.0f);
      ovlacc[r] = fmaxf(ovlacc[r], nonself * ov * pv);

      // both angles in [0,pi): mod(a_i - a_j, pi) is a conditional add
      float ad = rA[r] - c0.w;
      ad = (ad < 0.0f) ? ad + PI_F : ad;
      ad = fminf(ad, PI_F - ad);
      float al = fminf(fmaxf(2.0f * ad * (1.0f / (0.5f * PI_F)) - 1.0f, -1.0f), 1.0f);
      al = nonself * ((dm < MAXD) ? 1.0f : 0.0f) * al;
      alnacc[r] = fminf(alnacc[r], al);
    }
  }

  // reduce the 16 columns per half, then across waves via LDS
#pragma unroll
  for (int r = 0; r < 8; ++r) {
    ovlacc[r] = wave16_max(ovlacc[r]);
    alnacc[r] = wave16_min(alnacc[r]);
  }
  if (idx == 0) {
#pragma unroll
    for (int r = 0; r < 8; ++r) {
      redO[wave * 16 + m0 + r] = ovlacc[r];
      redA[wave * 16 + m0 + r] = alnacc[r];
    }
  }
  __syncthreads();
  if (tid < 16) {
    float o = 0.0f, a2 = 1e30f;
#pragma unroll
    for (int w2 = 0; w2 < NWAVES; ++w2) {
      o  = fmaxf(o,  redO[w2 * 16 + tid]);
      a2 = fminf(a2, redA[w2 * 16 + tid]);
    }
    ovlOut[b * P + i0 + tid] = o;
    alnOut[b * P + i0 + tid] = a2;
  }
}

// --------------------------- kernel 3: finalize ----------------------------
__global__ __launch_bounds__(256) void finalize_kernel(
    const float* __restrict__ peA, const float* __restrict__ weA,
    const float* __restrict__ leA, const float* __restrict__ aeA,
    const float* __restrict__ arA, const float* __restrict__ ovlA,
    const float* __restrict__ alnA, const float* __restrict__ ctxM,
    const float* __restrict__ comb_w, const float* __restrict__ comb_b,
    float* __restrict__ out, int B, int G, int P) {
  __shared__ float red[256];
  int b = blockIdx.x;
  float w0 = comb_w[0], w1 = comb_w[1], w2 = comb_w[2], w3 = comb_w[3];
  float w4 = comb_w[4], w5 = comb_w[5], w6 = comb_w[6];
  float bb = comb_b[0];
  float acc = 0.0f;
  for (int p = threadIdx.x; p < P; p += blockDim.x) {
    int i = b * P + p;
    float pm = ctxM[b * G + p];
    float e = w0 * peA[i] + w1 * weA[i] + w2 * leA[i] + w3 * aeA[i] +
              w4 * ovlA[i] + w5 * alnA[i] + w6 * arA[i] + bb;
    e *= pm;
    out[i] = e;
    acc += e;
  }
  red[threadIdx.x] = acc;
  __syncthreads();
  for (int s = 128; s > 0; s >>= 1) {
    if (threadIdx.x < (unsigned)s) red[threadIdx.x] += red[threadIdx.x + s];
    __syncthreads();
  }
  if (threadIdx.x == 0) out[(size_t)B * P + b] = red[0];
}

__global__ void total_kernel(float* __restrict__ out, int B, int P) {
  if (threadIdx.x == 0 && blockIdx.x == 0) {
    float t = 0.0f;
    for (int b = 0; b < B; ++b) t += out[(size_t)B * P + b];
    out[(size_t)B * P + B] = t;
  }
}

// ------------------------------ launcher -----------------------------------
extern "C" void kernel_launch(void* const* d_in, const int* in_sizes, int n_in,
                              void* d_out, int out_size, void* d_ws, size_t ws_size,
                              hipStream_t stream) {
  const float*         cc     = (const float*)d_in[0];
  const unsigned char* mask   = (const unsigned char*)d_in[1];
  const float*         pem    = (const float*)d_in[2];
  const float*         mwm    = (const float*)d_in[3];
  const float*         mlm    = (const float*)d_in[4];
  const float*         mam    = (const float*)d_in[5];
  const float*         comb_w = (const float*)d_in[6];
  const float*         comb_b = (const float*)d_in[7];
  float*               out    = (float*)d_out;

  // Derive shapes: in_sizes[1] = B*9N; out_size = B*P + B + 1 with P = N or 9N.
  int S1 = in_sizes[1];
  int B = out_size - 1 - S1 / 9;              // compute_context == 0 case
  if (B <= 0 || (out_size - 1) % B != 0) B = out_size - 1 - S1;  // fallback: P = 9N
  int P = (out_size - 1) / B - 1;
  int G = S1 / B;                              // 9N context points per batch

  size_t BG = (size_t)B * G, BP = (size_t)B * P;
  char* wsb = (char*)d_ws;
  float4* pk0 = (float4*)wsb;                 // BG float4
  float4* pk1 = pk0 + BG;                     // BG float4
  float*  ctxM = (float*)(pk1 + BG);          // BG floats
  float*  peA  = ctxM + BG;                   // 7 x BP floats
  float*  weA  = peA + BP;
  float*  leA  = weA + BP;
  float*  aeA  = leA + BP;
  float*  arA  = aeA + BP;
  float*  ovlA = arA + BP;
  float*  alnA = ovlA + BP;

  int nPts = B * G;
  prep_kernel<<<(nPts + 255) / 256, 256, 0, stream>>>(
      cc, mask, pem, mwm, mlm, mam,
      pk0, pk1, ctxM, peA, weA, leA, aeA, arA, B, G, P);

  pair_kernel<<<B * (P / 16), 256, 0, stream>>>(
      pk0, pk1, ctxM, ovlA, alnA, B, G, P);

  finalize_kernel<<<B, 256, 0, stream>>>(
      peA, weA, leA, aeA, arA, ovlA, alnA, ctxM, comb_w, comb_b, out, B, G, P);

  total_kernel<<<1, 32, 0, stream>>>(out, B, P);
}